// CustomSAGEConv_28346784153658
// MI455X (gfx1250) — compile-verified
//
#include <hip/hip_runtime.h>

typedef __attribute__((ext_vector_type(2))) float v2f;
typedef __attribute__((ext_vector_type(8))) float v8f;

#define N_NODES 50000
#define N_EDGES 800000
#define D 64
#define LDSW 66                       // padded LDS pitch to dodge bank conflicts
#define TILES ((N_NODES + 15) / 16)   // 3125 row tiles of 16

// ---------------------------------------------------------------------------
// Kernel 1: dual projection via V_WMMA_F32_16X16X4_F32.
//   msgs    = x @ W_msg^T  + b_msg   -> workspace
//   selfout = x @ W_self^T + b_self  -> d_out
// One wave per 16-row tile; 4 waves (128 threads) per block.
// ---------------------------------------------------------------------------
__global__ __launch_bounds__(128)
void sage_gemm_wmma(const float* __restrict__ x,
                    const float* __restrict__ Wm, const float* __restrict__ bm,
                    const float* __restrict__ Ws, const float* __restrict__ bs,
                    float* __restrict__ msgs, float* __restrict__ selfout)
{
    __shared__ float sWm[D * LDSW];
    __shared__ float sWs[D * LDSW];
    __shared__ float sBm[D];
    __shared__ float sBs[D];

    const int tid = threadIdx.x;
    for (int i = tid; i < D * D; i += 128) {
        const int r = i >> 6, k = i & 63;
        sWm[r * LDSW + k] = Wm[i];
        sWs[r * LDSW + k] = Ws[i];
    }
    if (tid < D) { sBm[tid] = bm[tid]; sBs[tid] = bs[tid]; }
    __syncthreads();

    const int wave = tid >> 5;
    const int lane = tid & 31;
    const int tile = blockIdx.x * 4 + wave;
    if (tile >= TILES) return;        // wave-uniform; EXEC stays all-ones

    const int half = lane >> 4;       // 0 -> K pair {0,1}; 1 -> K pair {2,3}
    const int lrow = lane & 15;       // A: matrix row M; B/C/D: column N (mod 16)
    const int row0 = tile * 16;

    v8f accM[4], accS[4];
#pragma unroll
    for (int t = 0; t < 4; ++t) {
        accM[t] = v8f{0.f, 0.f, 0.f, 0.f, 0.f, 0.f, 0.f, 0.f};
        accS[t] = v8f{0.f, 0.f, 0.f, 0.f, 0.f, 0.f, 0.f, 0.f};
    }

    const float* xrow = x + (size_t)(row0 + lrow) * D;

#pragma unroll 4
    for (int k0 = 0; k0 < D; k0 += 4) {
        const int ka = k0 + 2 * half;
        v2f a;                         // A 16x4: lane row = lrow, K = ka, ka+1
        a.x = xrow[ka];
        a.y = xrow[ka + 1];
#pragma unroll
        for (int t = 0; t < 4; ++t) {
            const int n = t * 16 + lrow;      // B column; B[k][n] = W[n][k]
            v2f bMf, bSf;
            bMf.x = sWm[n * LDSW + ka];  bMf.y = sWm[n * LDSW + ka + 1];
            bSf.x = sWs[n * LDSW + ka];  bSf.y = sWs[n * LDSW + ka + 1];
            accM[t] = __builtin_amdgcn_wmma_f32_16x16x4_f32(
                false, a, false, bMf, (short)0, accM[t], false, false);
            accS[t] = __builtin_amdgcn_wmma_f32_16x16x4_f32(
                false, a, false, bSf, (short)0, accS[t], false, false);
        }
    }

    // C/D layout: VGPR j holds M = j + 8*half, N = lrow (+ 16*t)
#pragma unroll
    for (int t = 0; t < 4; ++t) {
        const int n = t * 16 + lrow;
        const float biasM = sBm[n], biasS = sBs[n];
#pragma unroll
        for (int j = 0; j < 8; ++j) {
            const int m = j + 8 * half;
            const size_t off = (size_t)(row0 + m) * D + n;
            msgs[off]    = accM[t][j] + biasM;
            selfout[off] = accS[t][j] + biasS;
        }
    }
}

// ---------------------------------------------------------------------------
// Kernel 0: zero scatter accumulators (summed + counts) every launch.
// ---------------------------------------------------------------------------
__global__ void sage_zero(float* __restrict__ p, int n)
{
    const int i = blockIdx.x * blockDim.x + threadIdx.x;
    if (i < n) p[i] = 0.0f;
}

// ---------------------------------------------------------------------------
// Kernel 2: edge scatter. thread = (edge, dim). Coalesced gather from
// msgs[row], GLOBAL_ATOMIC_ADD_F32 into summed[col]; 1 count atomic per edge.
// ---------------------------------------------------------------------------
__global__ __launch_bounds__(256)
void sage_scatter(const int* __restrict__ ei, const float* __restrict__ msgs,
                  float* __restrict__ summed, float* __restrict__ counts)
{
    const long long gid = (long long)blockIdx.x * blockDim.x + threadIdx.x;
    const long long e = gid >> 6;
    if (e >= N_EDGES) return;
    const int d   = (int)(gid & 63);
    const int src = ei[e];                 // edge_index[0][e]
    const int dst = ei[N_EDGES + e];       // edge_index[1][e]
    const float v = msgs[(size_t)src * D + d];
    atomicAdd(&summed[(size_t)dst * D + d], v);
    if (d == 0) atomicAdd(&counts[dst], 1.0f);
}

// ---------------------------------------------------------------------------
// Kernel 3: out += summed / max(count, 1)   (self projection already in out)
// ---------------------------------------------------------------------------
__global__ __launch_bounds__(256)
void sage_finalize(const float* __restrict__ summed,
                   const float* __restrict__ counts, float* __restrict__ out)
{
    const int i = blockIdx.x * blockDim.x + threadIdx.x;
    if (i < N_NODES * D) {
        const int n = i >> 6;
        const float r = 1.0f / fmaxf(counts[n], 1.0f);
        out[i] += summed[i] * r;
    }
}

extern "C" void kernel_launch(void* const* d_in, const int* in_sizes, int n_in,
                              void* d_out, int out_size, void* d_ws, size_t ws_size,
                              hipStream_t stream)
{
    const float* x  = (const float*)d_in[0];
    const int*   ei = (const int*)  d_in[1];   // int32 [2][E]
    const float* Wm = (const float*)d_in[2];
    const float* bm = (const float*)d_in[3];
    const float* Ws = (const float*)d_in[4];
    const float* bs = (const float*)d_in[5];
    float* out = (float*)d_out;

    float* msgs   = (float*)d_ws;                        // [N, 64]
    float* summed = msgs   + (size_t)N_NODES * D;        // [N, 64]
    float* counts = summed + (size_t)N_NODES * D;        // [N]   (contiguous)

    const int nz = N_NODES * D + N_NODES;                // summed + counts
    sage_zero<<<(nz + 255) / 256, 256, 0, stream>>>(summed, nz);

    sage_gemm_wmma<<<(TILES + 3) / 4, 128, 0, stream>>>(x, Wm, bm, Ws, bs,
                                                        msgs, out);

    const long long nscatter = (long long)N_EDGES * D;
    sage_scatter<<<(unsigned)((nscatter + 255) / 256), 256, 0, stream>>>(
        ei, msgs, summed, counts);

    sage_finalize<<<(N_NODES * D + 255) / 256, 256, 0, stream>>>(summed, counts,
                                                                 out);
}